// O3TensorProduct_23905787969629
// MI455X (gfx1250) — compile-verified
//
#include <hip/hip_runtime.h>

typedef __attribute__((ext_vector_type(2))) float v2f;
typedef __attribute__((ext_vector_type(4))) float v4f;
typedef __attribute__((ext_vector_type(8))) float v8f;
typedef __attribute__((ext_vector_type(3), aligned(4))) float v3f;

// constants
#define PW0f        0.08838834764831845f   // sqrt(1/128)
#define KDf         0.051031036307982884f  // PW0 / sqrt(3)
#define C1f         0.08838834764831845f   // PW1 / sqrt(3) == sqrt(1/128)
#define C2f         0.7071067811865476f    // PW2 / sqrt(6) * 8 == 1/sqrt(2)

__device__ __forceinline__ v8f wmma4(v2f a, v2f b, v8f c) {
    // D = A(16x4 f32) * B(4x16 f32) + C(16x16 f32)
    return __builtin_amdgcn_wmma_f32_16x16x4_f32(false, a, false, b, (short)0, c,
                                                 false, false);
}

__device__ __forceinline__ v2f loadB(const float* __restrict__ w, int r0, int col) {
    // B layout: VGPR0 = W[r0][col], VGPR1 = W[r0+1][col]; r0 already includes +2 for hi half
    v2f b;
    b.x = w[r0 * 64 + col];
    b.y = w[r0 * 64 + 64 + col];
    return b;
}

__global__ __launch_bounds__(256)
void o3tp_wmma_kernel(const float* __restrict__ x1, const float* __restrict__ x2,
                      const float* __restrict__ w_ss, const float* __restrict__ w_vv0,
                      const float* __restrict__ w_sv, const float* __restrict__ w_vs,
                      const float* __restrict__ w_vv1, const float* __restrict__ bias,
                      float* __restrict__ out, int nrows)
{
    const int lane = threadIdx.x & 31;
    const int wave = threadIdx.x >> 5;
    const int l = lane & 15;          // row-within-tile (A) / col-within-tile (B,C)
    const int h = lane >> 4;          // half-wave selector
    const int row0 = (blockIdx.x * 8 + wave) * 16;
    if (row0 >= nrows) return;        // uniform per wave: EXEC stays all-ones

    const int rowA = row0 + l;        // A-fragment row owned by this lane
    const float* __restrict__ x1r = x1 + (size_t)rowA * 256;

    // per-lane x2 vector of rowA (for building the 'dot' A fragments)
    v4f x2rA = *(const v4f*)(x2 + (size_t)rowA * 4);
    const float vax = x2rA.y, vay = x2rA.z, vaz = x2rA.w;

    // ---- preload all A fragments (K = 0..63 in 16 steps of 4) into registers ----
    v2f Sf[16], V0f[16], V1f[16], V2f[16], Df[16];
#pragma unroll
    for (int ks = 0; ks < 16; ++ks) {
        const int k = 4 * ks + 2 * h;                 // this lane's K base
        Sf[ks] = *(const v2f*)(x1r + k);              // x1_s[row][k..k+1]
        v3f va = *(const v3f*)(x1r + 64 + 3 * k);     // x1_v[row][k][0..2]
        v3f vb = *(const v3f*)(x1r + 64 + 3 * k + 3); // x1_v[row][k+1][0..2]
        v2f t0, t1, t2;
        t0.x = va.x; t0.y = vb.x;
        t1.x = va.y; t1.y = vb.y;
        t2.x = va.z; t2.y = vb.z;
        V0f[ks] = t0; V1f[ks] = t1; V2f[ks] = t2;
        Df[ks] = t0 * vax + t1 * vay + t2 * vaz;      // dot[row][k..k+1]
    }

    // ---- per-output-row x2 values (C layout: VGPR j <-> row0 + j + 8h) ----
    float x2s_j[8], x2vx[8], x2vy[8], x2vz[8];
#pragma unroll
    for (int j = 0; j < 8; ++j) {
        v4f t = *(const v4f*)(x2 + (size_t)(row0 + j + 8 * h) * 4);
        x2s_j[j] = t.x; x2vx[j] = t.y; x2vy[j] = t.z; x2vz[j] = t.w;
    }

    // ---- N tiles of 16 columns ----
    for (int n = 0; n < 4; ++n) {
        const int col = n * 16 + l;
        v8f a_ss  = {0,0,0,0,0,0,0,0};
        v8f a_d   = {0,0,0,0,0,0,0,0};
        v8f a_sv  = {0,0,0,0,0,0,0,0};
        v8f a_vs0 = {0,0,0,0,0,0,0,0};
        v8f a_vs1 = {0,0,0,0,0,0,0,0};
        v8f a_vs2 = {0,0,0,0,0,0,0,0};
        v8f a_c0  = {0,0,0,0,0,0,0,0};
        v8f a_c1  = {0,0,0,0,0,0,0,0};
        v8f a_c2  = {0,0,0,0,0,0,0,0};

#pragma unroll
        for (int ks = 0; ks < 16; ++ks) {
            const int r0 = 4 * ks + 2 * h;
            const v2f Bss = loadB(w_ss,  r0, col);
            const v2f Bd0 = loadB(w_vv0, r0, col);
            const v2f Bsv = loadB(w_sv,  r0, col);
            const v2f Bvs = loadB(w_vs,  r0, col);
            const v2f Bv1 = loadB(w_vv1, r0, col);

            a_ss  = wmma4(Sf[ks],  Bss, a_ss);   // x1_s @ w_ss
            a_d   = wmma4(Df[ks],  Bd0, a_d);    // dot  @ w_vv0
            a_sv  = wmma4(Sf[ks],  Bsv, a_sv);   // x1_s @ w_sv
            a_vs0 = wmma4(V0f[ks], Bvs, a_vs0);  // V_i  @ w_vs
            a_vs1 = wmma4(V1f[ks], Bvs, a_vs1);
            a_vs2 = wmma4(V2f[ks], Bvs, a_vs2);
            a_c0  = wmma4(V0f[ks], Bv1, a_c0);   // V_i  @ w_vv1
            a_c1  = wmma4(V1f[ks], Bv1, a_c1);
            a_c2  = wmma4(V2f[ks], Bv1, a_c2);
        }

        const float bv = bias[col];

#pragma unroll
        for (int j = 0; j < 8; ++j) {
            const int row = row0 + j + 8 * h;
            float* __restrict__ orow = out + (size_t)row * 448;
            const float s2 = x2s_j[j];
            const float bx = x2vx[j], by = x2vy[j], bz = x2vz[j];

            // out0 = PW0*(ss*x2_s + dot@w_vv0 / sqrt(3)) + bias
            orow[col] = PW0f * (a_ss[j] * s2) + KDf * a_d[j] + bv;

            // out1 = PW1/sqrt(3) * (sv * x2_v + vs_i * x2_s)
            const float sv = a_sv[j];
            v3f o1;
            o1.x = C1f * (sv * bx + a_vs0[j] * s2);
            o1.y = C1f * (sv * by + a_vs1[j] * s2);
            o1.z = C1f * (sv * bz + a_vs2[j] * s2);
            *(v3f*)(orow + 64 + 3 * col) = o1;

            // out2 = (PW2/sqrt(6))*8 * cross(a, x2_v)
            const float ax = a_c0[j], ay = a_c1[j], az = a_c2[j];
            v3f o2;
            o2.x = C2f * (ay * bz - az * by);
            o2.y = C2f * (az * bx - ax * bz);
            o2.z = C2f * (ax * by - ay * bx);
            *(v3f*)(orow + 256 + 3 * col) = o2;
        }
    }
}

extern "C" void kernel_launch(void* const* d_in, const int* in_sizes, int n_in,
                              void* d_out, int out_size, void* d_ws, size_t ws_size,
                              hipStream_t stream) {
    (void)n_in; (void)out_size; (void)d_ws; (void)ws_size;
    const float* x1    = (const float*)d_in[0];
    const float* x2    = (const float*)d_in[1];
    const float* w_ss  = (const float*)d_in[2];
    const float* w_vv0 = (const float*)d_in[3];
    const float* w_sv  = (const float*)d_in[4];
    const float* w_vs  = (const float*)d_in[5];
    const float* w_vv1 = (const float*)d_in[6];
    const float* bias  = (const float*)d_in[7];
    float* out = (float*)d_out;

    const int nrows = in_sizes[0] / 256;          // 262144
    const int blocks = (nrows + 127) / 128;       // 128 rows per block (8 waves x 16)
    o3tp_wmma_kernel<<<blocks, 256, 0, stream>>>(x1, x2, w_ss, w_vv0, w_sv,
                                                 w_vs, w_vv1, bias, out, nrows);
}